// GCN_9345848836262
// MI455X (gfx1250) — compile-verified
//
#include <hip/hip_runtime.h>
#include <hip/hip_bf16.h>

typedef float v2f __attribute__((ext_vector_type(2)));
typedef float v8f __attribute__((ext_vector_type(8)));

#define HDIM 64

// ---------------------------------------------------------------------------
// Degree / norm helpers
// ---------------------------------------------------------------------------
__global__ void gcn_init_deg(float* __restrict__ deg, int n) {
    int i = blockIdx.x * blockDim.x + threadIdx.x;
    if (i < n) deg[i] = 1.0f;            // self-loop contributes 1
}

__global__ void gcn_edge_deg(const int* __restrict__ dst, float* __restrict__ deg, int e) {
    int i = blockIdx.x * blockDim.x + threadIdx.x;
    if (i < e) atomicAdd(&deg[dst[i]], 1.0f);
}

__global__ void gcn_deg_to_dinv(float* __restrict__ deg, int n) {
    int i = blockIdx.x * blockDim.x + threadIdx.x;
    if (i < n) deg[i] = rsqrtf(deg[i]);  // deg >= 1 always (self loops)
}

// norm[e] = dinv[src[e]] * dinv[dst[e]]   (computed once, reused by both convs)
__global__ void gcn_edge_norm(const int* __restrict__ src, const int* __restrict__ dst,
                              const float* __restrict__ dinv, float* __restrict__ norm,
                              int e) {
    int i = blockIdx.x * blockDim.x + threadIdx.x;
    if (i < e) norm[i] = dinv[src[i]] * dinv[dst[i]];
}

// ---------------------------------------------------------------------------
// Dense GEMM via V_WMMA_F32_16X16X4_F32 with fused self-loop epilogue.
//   T[r, c]     = (RELU ? relu(A) : A)[r, :] @ W[:, c] + bias[c]
//   Hinit[r, c] = T[r, c] * dinv[r]^2      (self-loop term of the aggregation)
// One wave computes one 16x16 tile; block = 4 waves = 4 column tiles (H=64).
// A == Hinit is allowed (conv2 operates in place on h): every block reads only
// its own 16 A-rows, and the __syncthreads() orders those reads before any
// epilogue writes within the block.
// Lane mapping (32-bit operands, wave32):
//   A 16x4 : lane l: M = l&15, VGPR0 -> K = k0 + 2*(l>>4), VGPR1 -> K+1
//   B 4x16 : lane l: N = l&15, VGPR0 -> K = k0 + 2*(l>>4), VGPR1 -> K+1
//   C/D    : lane l, VGPR v: M = v + 8*(l>>4), N = l&15
// ---------------------------------------------------------------------------
template <int K, bool RELU>
__global__ void gcn_gemm_wmma(const float* __restrict__ A,
                              const float* __restrict__ W,
                              const float* __restrict__ bias,
                              const float* __restrict__ dinv,
                              float* __restrict__ T,
                              float* __restrict__ Hinit,
                              int nrows) {
    const int wave = threadIdx.x >> 5;
    const int lane = threadIdx.x & 31;
    const int row0 = blockIdx.x * 16;
    const int col0 = wave * 16;
    if (row0 >= nrows) return;           // uniform per block: EXEC stays full

    const int m    = lane & 15;          // M for A, N for B/D
    const int half = lane >> 4;          // 0 or 1
    const int kb   = half * 2;

    v8f c = {};
    const float* arow = A + (size_t)(row0 + m) * K;
#pragma unroll 4
    for (int k0 = 0; k0 < K; k0 += 4) {
        v2f a;
        a.x = arow[k0 + kb];
        a.y = arow[k0 + kb + 1];
        if (RELU) {
            a.x = fmaxf(a.x, 0.0f);
            a.y = fmaxf(a.y, 0.0f);
        }
        v2f b;
        b.x = W[(size_t)(k0 + kb) * HDIM + col0 + m];
        b.y = W[(size_t)(k0 + kb + 1) * HDIM + col0 + m];
        c = __builtin_amdgcn_wmma_f32_16x16x4_f32(
                /*neg_a=*/false, a, /*neg_b=*/false, b,
                /*c_mod=*/(short)0, c, /*reuse_a=*/false, /*reuse_b=*/false);
    }

    __syncthreads();                     // all A reads done before Hinit writes

    const float bv = bias[col0 + m];
#pragma unroll
    for (int v = 0; v < 8; ++v) {
        int row = row0 + v + half * 8;
        float val = c[v] + bv;
        float di  = dinv[row];
        size_t idx = (size_t)row * HDIM + col0 + m;
        T[idx]     = val;
        Hinit[idx] = val * di * di;
    }
}

// ---------------------------------------------------------------------------
// Edge aggregation: out[d] += t[s] * norm[e]    (out pre-seeded with self term)
// 64 threads per edge (feature-parallel); coalesced gather, L2-resident atomics.
// ---------------------------------------------------------------------------
__global__ void gcn_agg_edges(const int* __restrict__ src,
                              const int* __restrict__ dst,
                              const float* __restrict__ t,
                              const float* __restrict__ norm,
                              float* __restrict__ out, int e) {
    int tid  = blockIdx.x * blockDim.x + threadIdx.x;
    int edge = tid >> 6;
    int f    = tid & 63;
    if (edge >= e) return;
    int s = src[edge];
    int d = dst[edge];
    atomicAdd(&out[(size_t)d * HDIM + f], t[(size_t)s * HDIM + f] * norm[edge]);
}

// ---------------------------------------------------------------------------
// Global mean pool + output projection
// ---------------------------------------------------------------------------
__global__ void gcn_pool_init(float* __restrict__ sums, float* __restrict__ cnt,
                              int g) {
    int i = blockIdx.x * blockDim.x + threadIdx.x;
    if (i < g * HDIM) sums[i] = 0.0f;
    if (i < g) cnt[i] = 0.0f;
}

__global__ void gcn_pool_accum(const int* __restrict__ batch,
                               const float* __restrict__ h,
                               float* __restrict__ sums,
                               float* __restrict__ cnt, int n) {
    int tid  = blockIdx.x * blockDim.x + threadIdx.x;
    int node = tid >> 6;
    int f    = tid & 63;
    if (node >= n) return;
    int g = batch[node];
    atomicAdd(&sums[(size_t)g * HDIM + f], h[(size_t)node * HDIM + f]);
    if (f == 0) atomicAdd(&cnt[g], 1.0f);
}

__global__ void gcn_final(const float* __restrict__ sums,
                          const float* __restrict__ cnt,
                          const float* __restrict__ Wout,
                          const float* __restrict__ bout,
                          float* __restrict__ out, int g, int c) {
    int tid = blockIdx.x * blockDim.x + threadIdx.x;
    if (tid >= g * c) return;
    int gg = tid / c;
    int cc = tid - gg * c;
    float inv = 1.0f / fmaxf(cnt[gg], 1.0f);
    float acc = bout[cc];
#pragma unroll 8
    for (int k = 0; k < HDIM; ++k)
        acc = fmaf(sums[(size_t)gg * HDIM + k] * inv, Wout[k * c + cc], acc);
    out[tid] = acc;
}

// ---------------------------------------------------------------------------
// Host-side launcher
// ---------------------------------------------------------------------------
extern "C" void kernel_launch(void* const* d_in, const int* in_sizes, int n_in,
                              void* d_out, int out_size, void* d_ws, size_t ws_size,
                              hipStream_t stream) {
    const float* x     = (const float*)d_in[0];   // [N, F_IN]
    const int*   eidx  = (const int*)d_in[1];     // [2, E]
    const int*   batch = (const int*)d_in[2];     // [N]
    const float* W1    = (const float*)d_in[3];   // [F_IN, 64]
    const float* b1    = (const float*)d_in[4];   // [64]
    const float* W2    = (const float*)d_in[5];   // [64, 64]
    const float* b2    = (const float*)d_in[6];   // [64]
    const float* Wout  = (const float*)d_in[7];   // [64, C]
    const float* bout  = (const float*)d_in[8];   // [C]
    float*       out   = (float*)d_out;           // [G, C]

    const int N    = in_sizes[2];
    const int E    = in_sizes[1] / 2;
    const int FIN  = in_sizes[0] / N;             // 128
    const int C    = in_sizes[8];                 // 19
    const int G    = out_size / C;                // 512
    (void)FIN; (void)n_in; (void)ws_size;

    const int* src = eidx;        // row 0
    const int* dst = eidx + E;    // row 1

    // workspace layout (floats): ~33 MB total
    float* ws   = (float*)d_ws;
    float* dinv = ws;                              // N    (deg, then dinv)
    float* norm = dinv + N;                        // E    (per-edge norm)
    float* t    = norm + E;                        // N*64 (pre-aggregation features)
    float* h    = t + (size_t)N * HDIM;            // N*64 (aggregated features)
    float* sums = h + (size_t)N * HDIM;            // G*64
    float* cnt  = sums + (size_t)G * HDIM;         // G

    const int TB = 256;
    dim3 blk(TB);

    // degree -> dinv -> per-edge norm (norm reused by both convolutions)
    gcn_init_deg<<<(N + TB - 1) / TB, blk, 0, stream>>>(dinv, N);
    gcn_edge_deg<<<(E + TB - 1) / TB, blk, 0, stream>>>(dst, dinv, E);
    gcn_deg_to_dinv<<<(N + TB - 1) / TB, blk, 0, stream>>>(dinv, N);
    gcn_edge_norm<<<(E + TB - 1) / TB, blk, 0, stream>>>(src, dst, dinv, norm, E);

    // conv1: t = x@W1+b1, h seeded with self-loop term, then edge scatter-add
    gcn_gemm_wmma<128, false><<<(N + 15) / 16, dim3(128), 0, stream>>>(x, W1, b1, dinv, t, h, N);
    gcn_agg_edges<<<((size_t)E * HDIM + TB - 1) / TB, blk, 0, stream>>>(src, dst, t, norm, h, E);

    // conv2: t = relu(h)@W2+b2 (relu fused in A-load), h re-seeded in place
    gcn_gemm_wmma<64, true><<<(N + 15) / 16, dim3(128), 0, stream>>>(h, W2, b2, dinv, t, h, N);
    gcn_agg_edges<<<((size_t)E * HDIM + TB - 1) / TB, blk, 0, stream>>>(src, dst, t, norm, h, E);

    // mean pool + output projection
    gcn_pool_init<<<((size_t)G * HDIM + TB - 1) / TB, blk, 0, stream>>>(sums, cnt, G);
    gcn_pool_accum<<<((size_t)N * HDIM + TB - 1) / TB, blk, 0, stream>>>(batch, h, sums, cnt, N);
    gcn_final<<<(G * C + TB - 1) / TB, blk, 0, stream>>>(sums, cnt, Wout, bout, out, G, C);
}